// TypedLatentClassifier_69123203662022
// MI455X (gfx1250) — compile-verified
//
#include <hip/hip_runtime.h>

// TypedLatentClassifier — MI455X (gfx1250) streaming token-scan kernel.
// Bandwidth-bound: 64 MB token stream read once -> ~2.9 us floor @ 23.3 TB/s.
// Data path: global_load_async_to_lds_b128 (ASYNCcnt) -> LDS (ds_load_b128) ->
// branchless per-lane scan -> wave32 shuffle segmented-scan carry ->
// LDS atomicOr reduce -> 8-float store per row.

#define S_LEN  4096
#define NTHR   256
#define TPT    16          // tokens per thread (S_LEN / NTHR)
#define NWAVE  (NTHR / 32)

typedef int v4i __attribute__((ext_vector_type(4)));
typedef __attribute__((address_space(1))) v4i as1_v4i;
typedef __attribute__((address_space(3))) v4i as3_v4i;
typedef __attribute__((address_space(3))) char as3_char;

__device__ __forceinline__ unsigned umaxu(unsigned a, unsigned b) { return a > b ? a : b; }

__global__ __launch_bounds__(NTHR) void tlc_kernel(const int* __restrict__ tok_ids,
                                                   const float* __restrict__ lut,
                                                   float* __restrict__ out)
{
    __shared__ int      toks[S_LEN + 4];   // +sentinel slot at [S_LEN]
    __shared__ unsigned wavemax[NWAVE];
    __shared__ unsigned smask[3];

    const int tid  = threadIdx.x;
    const int row  = blockIdx.x;
    const int lane = tid & 31;
    const int wave = tid >> 5;

    if (tid < 3)        smask[tid] = 0u;
    if (tid == NTHR - 1) toks[S_LEN] = 0;  // sentinel: not a tag, not a role, never a value

    // ---- stage row into LDS via CDNA5 async global->LDS copies ----
    const int* grow = tok_ids + (size_t)row * S_LEN;
    #pragma unroll
    for (int i = 0; i < S_LEN / (NTHR * 4); ++i) {       // 4 ints per b128
        const int idx = (i * NTHR + tid) * 4;
        const int* g  = grow + idx;
        int* l        = &toks[idx];
#if __has_builtin(__builtin_amdgcn_global_load_async_to_lds_b128)
        __builtin_amdgcn_global_load_async_to_lds_b128(
            (as1_v4i*)(void*)g, (as3_v4i*)(void*)l, /*offset=*/0, /*cpol=*/0);
#else
        unsigned laddr = (unsigned)(unsigned long long)(as3_char*)(void*)l;
        asm volatile("global_load_async_to_lds_b128 %0, %1, off"
                     :: "v"(laddr), "v"(g) : "memory");
#endif
    }
#if __has_builtin(__builtin_amdgcn_s_wait_asynccnt)
    __builtin_amdgcn_s_wait_asynccnt(0);
#else
    asm volatile("s_wait_asynccnt 0" ::: "memory");
#endif
    __syncthreads();

    // ---- pull this thread's 16 tokens (+1 boundary) from LDS, vectorized ----
    const int base = tid * TPT;
    int t[TPT + 1];
    #pragma unroll
    for (int k = 0; k < TPT / 4; ++k) {
        const int4 q = *(const int4*)&toks[base + 4 * k];
        t[4 * k + 0] = q.x; t[4 * k + 1] = q.y; t[4 * k + 2] = q.z; t[4 * k + 3] = q.w;
    }
    t[TPT] = toks[base + TPT];             // sentinel-safe for the last thread

    // ---- branchless local scan ----
    // acc[r]: bits[23:0] = value bits for tags 0..2 (8 bits each),
    //         bits[31:24] = "pending" (matched before any local tag).
    unsigned sh = 24u;                     // 8*latestTag, 24 == no tag yet
    unsigned acc0 = 0u, acc1 = 0u, acc2 = 0u;

    #pragma unroll
    for (int j = 0; j < TPT; ++j) {
        const int c = t[j];
        const int n = t[j + 1];
        const unsigned tg = (unsigned)(c - 46);          // TAG_START
        sh = (tg < 3u) ? (tg << 3) : sh;                 // update latest-tag shift
        const unsigned r  = (unsigned)(c - 3);           // ROLE_A..ROLE_C -> 0..2
        const unsigned v  = (unsigned)(n - 22) - (r << 3); // n - (22 + 8r)
        const bool hit    = (r < 3u) & (v < 8u);
        const unsigned sb = hit ? (1u << (v + sh)) : 0u; // v<8, sh<=24 -> fits 32b
        acc0 |= (r == 0u) ? sb : 0u;
        acc1 |= (r == 1u) ? sb : 0u;
        acc2 |= (r == 2u) ? sb : 0u;
    }

    // ---- resolve incoming latest-tag: wave32 max-scan + cross-wave combine ----
    // key encodes (position, tag); monotone in tid so max == most recent tag.
    const unsigned key = (sh != 24u) ? (unsigned)(tid * 4 + (sh >> 3) + 1) : 0u;
    unsigned incl = key;
    #pragma unroll
    for (int d = 1; d < 32; d <<= 1) {
        unsigned o = __shfl_up(incl, d, 32);
        if (lane >= d) incl = umaxu(incl, o);
    }
    unsigned excl = __shfl_up(incl, 1, 32);
    if (lane == 0)  excl = 0u;
    if (lane == 31) wavemax[wave] = incl;
    __syncthreads();

    unsigned prior = excl;
    for (int w = 0; w < wave; ++w) prior = umaxu(prior, wavemax[w]);

    unsigned m0 = acc0 & 0x00FFFFFFu;
    unsigned m1 = acc1 & 0x00FFFFFFu;
    unsigned m2 = acc2 & 0x00FFFFFFu;
    if (prior > 0u) {                      // attribute pending matches to carry-in tag
        const int isft = (int)((prior - 1u) & 3u) * 8;
        m0 |= (acc0 >> 24) << isft;
        m1 |= (acc1 >> 24) << isft;
        m2 |= (acc2 >> 24) << isft;
    }
    if (m0) atomicOr(&smask[0], m0);
    if (m1) atomicOr(&smask[1], m1);
    if (m2) atomicOr(&smask[2], m2);
    __syncthreads();

    // ---- final per-row classification (trivial; thread 0) ----
    if (tid == 0) {
        int qt = toks[S_LEN - 1] - 46;  qt = qt < 0 ? 0 : (qt > 2 ? 2 : qt);
        int ti = toks[1] - 49;          ti = ti < 0 ? 0 : (ti > 3 ? 3 : ti);

        const unsigned ma = (smask[0] >> (qt * 8)) & 0xFFu;
        const unsigned mb = (smask[1] >> (qt * 8)) & 0xFFu;
        const unsigned mc = (smask[2] >> (qt * 8)) & 0xFFu;
        // argmax over {0,1} floats == lowest set bit, else 0
        const int ai = ma ? (__ffs(ma) - 1) : 0;
        const int bi = mb ? (__ffs(mb) - 1) : 0;
        const int ci = mc ? (__ffs(mc) - 1) : 0;

        float res[8];
        if (ti == 3) {
            const float4* rowp = (const float4*)(lut + (((ai * 8 + bi) * 8 + ci) * 8));
            float4 lo = rowp[0], hi = rowp[1];
            res[0] = lo.x; res[1] = lo.y; res[2] = lo.z; res[3] = lo.w;
            res[4] = hi.x; res[5] = hi.y; res[6] = hi.z; res[7] = hi.w;
        } else {
            const int idx = (ti == 0) ? ai
                          : (ti == 1) ? ((ai + 2 * bi + 3 * ci) & 7)
                                      : ((ai * (bi + 1) + ci * ((ai ^ bi) + 1)) & 7);
            #pragma unroll
            for (int k = 0; k < 8; ++k) res[k] = 0.0f;
            res[idx] = 12.0f;
        }
        float4* o = (float4*)(out + (size_t)row * 8);
        o[0] = make_float4(res[0], res[1], res[2], res[3]);
        o[1] = make_float4(res[4], res[5], res[6], res[7]);
    }
}

extern "C" void kernel_launch(void* const* d_in, const int* in_sizes, int n_in,
                              void* d_out, int out_size, void* d_ws, size_t ws_size,
                              hipStream_t stream) {
    (void)n_in; (void)out_size; (void)d_ws; (void)ws_size;
    const int*   tok = (const int*)d_in[0];
    const float* lut = (const float*)d_in[1];
    float*       out = (float*)d_out;
    const int rows = in_sizes[0] / S_LEN;   // B = 4096
    tlc_kernel<<<rows, NTHR, 0, stream>>>(tok, lut, out);
}